// ValueToCoeff_71554155151983
// MI455X (gfx1250) — compile-verified
//
#include <hip/hip_runtime.h>

// ---------------------------------------------------------------------------
// Cubic B-spline prefilter on (4,2,192,192,192) f32, recast as three dense
// 192x192 matrix applications so the work runs on V_WMMA_F32_16X16X4_F32.
// ---------------------------------------------------------------------------

typedef float v2f __attribute__((ext_vector_type(2)));
typedef float v8f __attribute__((ext_vector_type(8)));

#define N_FILT 192
#define KSTEPS 48          // 192 / 4
#define ITILES 12          // 192 / 16
#define IPAIRS 6           // 192 / 32

// ---------------------------------------------------------------------------
// Build M (192x192): column j = exact reference IIR applied to 6*e_j, in f64.
// ---------------------------------------------------------------------------
__global__ void init_M_kernel(float* __restrict__ M) {
    const int j = threadIdx.x;
    if (j >= N_FILT) return;
    const int n = N_FILT;
    const double z = -0.2679491924311227065;   // sqrt(3) - 2

    // w[j] = (z^(j+1) + z^(2n-j)) / (1 - z^(2n))
    double zj1 = 1.0;
    for (int t = 0; t < j + 1; ++t) zj1 *= z;
    double z2nj = 1.0;
    for (int t = 0; t < 2 * n - j; ++t) z2nj *= z;
    double z2n = 1.0;
    for (int t = 0; t < 2 * n; ++t) z2n *= z;
    const double w = (zj1 + z2nj) / (1.0 - z2n);

    double y[N_FILT];
    // causal init: c0 = x[0] + sum_k w[k] x[k], with x = 6*e_j
    double c = ((j == 0) ? 6.0 : 0.0) + 6.0 * w;
    y[0] = c;
    for (int i = 1; i < n; ++i) {
        const double xi = (i == j) ? 6.0 : 0.0;
        c = xi + z * c;
        y[i] = c;
    }
    // anticausal init: c(n-1) = z/(z-1) * y(n-1)
    double o = y[n - 1] * (z / (z - 1.0));
    M[(n - 1) * n + j] = (float)o;
    for (int i = n - 2; i >= 0; --i) {
        o = z * (o - y[i]);
        M[i * n + j] = (float)o;
    }
}

// ---------------------------------------------------------------------------
// Repack M into the 16x4 f32 A-operand lane layout:
// lane l<16 : (M[row, k4*4+0], M[row, k4*4+1]);  lane>=16 : K offsets 2,3.
// Aprep[(itile*48 + k4)*32 + lane] = float2
// ---------------------------------------------------------------------------
__global__ void init_Aprep_kernel(const float* __restrict__ M, float* __restrict__ Aprep) {
    const int lane  = threadIdx.x;          // 0..31
    const int tile  = blockIdx.x;           // 0..575 = itile*48 + k4
    const int itile = tile / KSTEPS;
    const int k4    = tile % KSTEPS;
    const int row   = itile * 16 + (lane & 15);
    const int kk    = (lane < 16) ? 0 : 2;
    const int c     = k4 * 4 + kk;
    float* dst = Aprep + ((long)tile * 32 + lane) * 2;
    dst[0] = M[row * N_FILT + c];
    dst[1] = M[row * N_FILT + c + 1];
}

// ---------------------------------------------------------------------------
// Strided pass (inner size S, S multiple of 64):
//   dst[o, i, s] = sum_j M[i, j] * src[o, j, s]
// One wave -> C macro-tile 32(i) x 64(s) = 8 accumulators, 48 K-steps.
// The same 8 B loads per K-step now feed 8 WMMAs (B reuse across 2 i-tiles).
// ---------------------------------------------------------------------------
__global__ __launch_bounds__(256) void pass_strided_kernel(
        const float* __restrict__ src, float* __restrict__ dst,
        const float* __restrict__ Aprep, int O, int S, int num_sgroups) {
    const int lane = threadIdx.x & 31;
    const int wave = threadIdx.x >> 5;
    long idx = (long)blockIdx.x * 8 + wave;

    const int sg = (int)(idx % num_sgroups);
    idx /= num_sgroups;
    const int ipair = (int)(idx % IPAIRS);
    const int o = (int)(idx / IPAIRS);
    if (o >= O) return;

    const int itile0 = ipair * 2;
    const int s0     = sg * 64;
    const int nhalf  = lane & 15;
    const int krow   = (lane < 16) ? 0 : 2;

    v8f c00 = {}, c01 = {}, c02 = {}, c03 = {};   // i-tile 0, 4 s-tiles
    v8f c10 = {}, c11 = {}, c12 = {}, c13 = {};   // i-tile 1, 4 s-tiles
    const float* ap0 = Aprep + (((long)itile0 * KSTEPS) * 32 + lane) * 2;
    const float* ap1 = ap0 + (long)KSTEPS * 32 * 2;
    const long base = (long)o * N_FILT * S;

    for (int k4 = 0; k4 < KSTEPS; ++k4) {
        const v2f a0 = *(const v2f*)ap0;
        const v2f a1 = *(const v2f*)ap1;
        ap0 += 64; ap1 += 64;                   // 32 lanes * 2 floats
        const int j = k4 * 4 + krow;
        const float* r0 = src + base + (long)j * S + s0 + nhalf;
        const float* r1 = r0 + S;               // K row j+1
        v2f b0, b1, b2, b3;
        b0.x = r0[0];  b0.y = r1[0];
        b1.x = r0[16]; b1.y = r1[16];
        b2.x = r0[32]; b2.y = r1[32];
        b3.x = r0[48]; b3.y = r1[48];
        c00 = __builtin_amdgcn_wmma_f32_16x16x4_f32(false, a0, false, b0, (short)0, c00, false, false);
        c01 = __builtin_amdgcn_wmma_f32_16x16x4_f32(false, a0, false, b1, (short)0, c01, false, false);
        c02 = __builtin_amdgcn_wmma_f32_16x16x4_f32(false, a0, false, b2, (short)0, c02, false, false);
        c03 = __builtin_amdgcn_wmma_f32_16x16x4_f32(false, a0, false, b3, (short)0, c03, false, false);
        c10 = __builtin_amdgcn_wmma_f32_16x16x4_f32(false, a1, false, b0, (short)0, c10, false, false);
        c11 = __builtin_amdgcn_wmma_f32_16x16x4_f32(false, a1, false, b1, (short)0, c11, false, false);
        c12 = __builtin_amdgcn_wmma_f32_16x16x4_f32(false, a1, false, b2, (short)0, c12, false, false);
        c13 = __builtin_amdgcn_wmma_f32_16x16x4_f32(false, a1, false, b3, (short)0, c13, false, false);
    }

    // C layout: VGPR v -> i = itile*16 + v (+8 for lanes 16..31), n = lane&15
    const int ih = (lane < 16) ? 0 : 8;
    const long rb0 = base + (long)(itile0 * 16) * S + s0 + nhalf;
    const long rb1 = rb0 + (long)16 * S;
#pragma unroll
    for (int v = 0; v < 8; ++v) {
        const long r0 = rb0 + (long)(v + ih) * S;
        dst[r0 + 0]  = c00[v];
        dst[r0 + 16] = c01[v];
        dst[r0 + 32] = c02[v];
        dst[r0 + 48] = c03[v];
        const long r1 = rb1 + (long)(v + ih) * S;
        dst[r1 + 0]  = c10[v];
        dst[r1 + 16] = c11[v];
        dst[r1 + 32] = c12[v];
        dst[r1 + 48] = c13[v];
    }
}

// ---------------------------------------------------------------------------
// Last (contiguous) axis pass via LDS transpose:
//   dst[l, i] = sum_j M[i, j] * src[l, j]
// Block = 256 threads, 32 lines staged in LDS (stride 196 to avoid bank
// conflicts). 24 C tiles per block, 3 per wave. Output goes back through LDS
// so global stores stay fully coalesced.
// ---------------------------------------------------------------------------
#define LDS_STRIDE 196
__global__ __launch_bounds__(256) void pass_lastaxis_kernel(
        const float* __restrict__ src, float* __restrict__ dst,
        const float* __restrict__ Aprep) {
    __shared__ float lds[32 * LDS_STRIDE];
    const int tid  = threadIdx.x;
    const int lane = tid & 31;
    const int wave = tid >> 5;
    const long flat0 = (long)blockIdx.x * 32 * N_FILT;

    // stage 32 contiguous lines (fully coalesced: flat = flat0 + idx)
#pragma unroll
    for (int p = 0; p < 24; ++p) {
        const int idx = p * 256 + tid;
        const int l = idx / N_FILT;
        const int e = idx % N_FILT;
        lds[l * LDS_STRIDE + e] = src[flat0 + idx];
    }
    __syncthreads();

    const int nhalf = lane & 15;
    const int krow  = (lane < 16) ? 0 : 2;

    v8f acc[3];
#pragma unroll
    for (int q = 0; q < 3; ++q) {
        const int t     = wave * 3 + q;       // 0..23
        const int ltile = t & 1;
        const int itile = t >> 1;
        v8f c = {};
        const float* ap   = Aprep + (((long)itile * KSTEPS) * 32 + lane) * 2;
        const float* lrow = lds + (ltile * 16 + nhalf) * LDS_STRIDE;
        for (int k4 = 0; k4 < KSTEPS; ++k4) {
            const v2f a = *(const v2f*)ap;
            ap += 64;
            const int j = k4 * 4 + krow;      // even -> 8B-aligned ds_load_b64
            const v2f b = *(const v2f*)(lrow + j);
            c = __builtin_amdgcn_wmma_f32_16x16x4_f32(false, a, false, b, (short)0, c, false, false);
        }
        acc[q] = c;
    }
    __syncthreads();

    // transpose C tiles back into LDS: lds[line * LDS_STRIDE + i]
#pragma unroll
    for (int q = 0; q < 3; ++q) {
        const int t     = wave * 3 + q;
        const int ltile = t & 1;
        const int itile = t >> 1;
        const int l     = ltile * 16 + nhalf;
        const int ibase = itile * 16 + ((lane < 16) ? 0 : 8);
#pragma unroll
        for (int v = 0; v < 8; ++v)
            lds[l * LDS_STRIDE + ibase + v] = acc[q][v];
    }
    __syncthreads();

    // coalesced store
#pragma unroll
    for (int p = 0; p < 24; ++p) {
        const int idx = p * 256 + tid;
        const int l = idx / N_FILT;
        const int e = idx % N_FILT;
        dst[flat0 + idx] = lds[l * LDS_STRIDE + e];
    }
}

// ---------------------------------------------------------------------------
// kernel_launch
//   ws layout: [0,147456) A-prep | [147456,294912) M | [294912,...) tmp tensor
//   needs ws_size >= 294912 + 4*56623104 bytes (~226.8 MB)
// ---------------------------------------------------------------------------
extern "C" void kernel_launch(void* const* d_in, const int* in_sizes, int n_in,
                              void* d_out, int out_size, void* d_ws, size_t ws_size,
                              hipStream_t stream) {
    (void)in_sizes; (void)n_in; (void)out_size; (void)ws_size;

    const float* x   = (const float*)d_in[0];
    float*       out = (float*)d_out;
    char*        ws  = (char*)d_ws;

    float* Aprep = (float*)ws;                       // 147,456 B
    float* M     = (float*)(ws + 147456);            // 147,456 B
    float* tmp   = (float*)(ws + 294912);            // 226,492,416 B

    // build the filter matrix + WMMA A-operand packing (L2-resident, 144 KB)
    init_M_kernel<<<1, 192, 0, stream>>>(M);
    init_Aprep_kernel<<<ITILES * KSTEPS, 32, 0, stream>>>(M, Aprep);

    // pass 1: axis 2 (z), inner S = 192*192, outer O = 4*2 = 8 ; x -> out
    {
        const int S = N_FILT * N_FILT, O = 8, sg = S / 64;
        const long waves = (long)O * IPAIRS * sg;    // 27,648
        pass_strided_kernel<<<(int)(waves / 8), 256, 0, stream>>>(x, out, Aprep, O, S, sg);
    }
    // pass 2: axis 3 (y), inner S = 192, outer O = 8*192 ; out -> tmp
    {
        const int S = N_FILT, O = 8 * N_FILT, sg = S / 64;
        const long waves = (long)O * IPAIRS * sg;    // 27,648
        pass_strided_kernel<<<(int)(waves / 8), 256, 0, stream>>>(out, tmp, Aprep, O, S, sg);
    }
    // pass 3: axis 4 (x), contiguous ; tmp -> out  (294,912 lines / 32 per block)
    pass_lastaxis_kernel<<<294912 / 32, 256, 0, stream>>>(tmp, out, Aprep);
}